// FixedPtSQP_Plus_70489003262100
// MI455X (gfx1250) — compile-verified
//
#include <hip/hip_runtime.h>

// FixedPtSQP_Plus for MI455X (gfx1250, wave32).
// One batch item per wave; 4 items per 128-thread workgroup.
// Batched Z@Q (fp32) uses V_WMMA_F32_16X16X4_F32 (full-precision WMMA).
// KKT solve: H = Q + diag (SPD) -> packed-lower Cholesky in LDS per wave,
// two triangular solves (u = H^-1 rhs, w = H^-1 1), scalar Schur for dnu.

#define NN 64
#define ITEMS 4
#define WG 128
#define NEWTON_IT 15
#define SQP_TOTAL 6
#define TRIN (NN * (NN + 1) / 2)

typedef float v2f __attribute__((ext_vector_type(2)));
typedef float v8f __attribute__((ext_vector_type(8)));

__device__ __forceinline__ int TRI(int r, int c) { return ((r * (r + 1)) >> 1) + c; }

__device__ __forceinline__ float wsum(float v) {
#pragma unroll
  for (int off = 16; off; off >>= 1) v += __shfl_xor(v, off, 32);
  return v;
}
__device__ __forceinline__ float wmin(float v) {
#pragma unroll
  for (int off = 16; off; off >>= 1) v = fminf(v, __shfl_xor(v, off, 32));
  return v;
}

// sZQ(16x64) = sZ(16x64) @ sQ(64x64), fp32 WMMA 16x16x4.
// Each of the 4 waves computes one 16-column strip. Rows >= ITEMS are zeroed
// in sZ once at startup; their outputs are never read.
__device__ __forceinline__ void gemm_ZQ(const float* sZ, const float* sQ,
                                        float* sZQ, int wave, int lane) {
  __syncthreads();  // all Z-row writes visible
  {
    const int n0 = (wave & 3) * 16;
    const int mn = lane & 15;   // A row (M) / B,C column (N)
    const int hi = lane >> 4;   // 0: lanes 0-15, 1: lanes 16-31
    v8f acc = {};
#pragma unroll
    for (int k0 = 0; k0 < NN; k0 += 4) {
      v2f a, b;
      // A 16x4 fp32: VGPR0 = K {0 | 2}, VGPR1 = K {1 | 3}
      a.x = sZ[mn * NN + k0 + hi * 2];
      a.y = sZ[mn * NN + k0 + hi * 2 + 1];
      // B 4x16 fp32: VGPR0 = rows {0 | 2}, VGPR1 = rows {1 | 3}
      b.x = sQ[(k0 + hi * 2) * NN + n0 + mn];
      b.y = sQ[(k0 + hi * 2 + 1) * NN + n0 + mn];
      acc = __builtin_amdgcn_wmma_f32_16x16x4_f32(false, a, false, b, (short)0,
                                                  acc, false, false);
    }
#pragma unroll
    for (int i = 0; i < 8; ++i)  // D: VGPR i = rows {i | i+8}
      sZQ[(i + hi * 8) * NN + n0 + mn] = acc[i];
  }
  __syncthreads();  // results visible before consumers read
}

__global__ __launch_bounds__(WG) void sqp_kernel(const float* __restrict__ cmat,
                                                 const float* __restrict__ Qmat,
                                                 float* __restrict__ out) {
  __shared__ float sQ[NN * NN];       // Q0, shared by all items
  __shared__ float sH[ITEMS * TRIN];  // packed lower H per item
  __shared__ float sZ[16 * NN];       // GEMM A tile (rows >= ITEMS zero)
  __shared__ float sZQ[16 * NN];      // GEMM D tile
  __shared__ float sU[ITEMS * NN];    // rhs -> u solve buffer
  __shared__ float sW[ITEMS * NN];    // ones -> w solve buffer

  const int tid = threadIdx.x;
  const int wave = tid >> 5;
  const int lane = tid & 31;
  const int gitem = blockIdx.x * ITEMS + wave;

  for (int i = tid; i < NN * NN; i += WG) sQ[i] = Qmat[i];
  for (int i = ITEMS * NN + tid; i < 16 * NN; i += WG) sZ[i] = 0.0f;
  __syncthreads();

  float* Hi = sH + wave * TRIN;
  float* Ui = sU + wave * NN;
  float* Wi = sW + wave * NN;
  float* Zr = sZ + wave * NN;
  float* ZQr = sZQ + wave * NN;

  // lane owns indices {lane, lane+32} of 64-vectors and
  // {lane, lane+32, lane+64, lane+96} of 128-vectors.
  const float c0 = cmat[gitem * NN + lane];
  const float c1 = cmat[gitem * NN + lane + 32];

  float x0 = 0.5f, x1 = 0.5f;
  float d0 = 0.0f, d1 = 0.0f;

  for (int it = 0; it < SQP_TOTAL; ++it) {
    // p = x @ Q0 - c  (WMMA)
    Zr[lane] = x0;
    Zr[lane + 32] = x1;
    gemm_ZQ(sZ, sQ, sZQ, wave, lane);
    const float p0 = ZQr[lane] - c0;
    const float p1 = ZQr[lane + 32] - c1;
    const float bb = 1.0f - wsum(x0 + x1);
    const float h0 = 1.0f - x0, h1 = 1.0f - x1, h2 = x0, h3 = x1;

    // QP state init: z=0, s=1, lam=1, nu=0
    float z0 = 0.0f, z1 = 0.0f;
    float s0 = 1.0f, s1 = 1.0f, s2 = 1.0f, s3 = 1.0f;
    float l0 = 1.0f, l1 = 1.0f, l2 = 1.0f, l3 = 1.0f;
    float nu = 0.0f;

    for (int nit = 0; nit < NEWTON_IT; ++nit) {
      // Qz (WMMA; Q symmetric so Z@Q rows are Q@z_i)
      Zr[lane] = z0;
      Zr[lane + 32] = z1;
      gemm_ZQ(sZ, sQ, sZQ, wave, lane);
      const float qz0 = ZQr[lane];
      const float qz1 = ZQr[lane + 32];

      const float mu =
          0.1f * wsum(s0 * l0 + s1 * l1 + s2 * l2 + s3 * l3) * (1.0f / 128.0f);

      // residuals (G = [I;-I], A = ones)
      const float rd0 = qz0 + p0 + (l0 - l2) + nu;
      const float rd1 = qz1 + p1 + (l1 - l3) + nu;
      const float rp0 = z0 + s0 - h0;
      const float rp1 = z1 + s1 - h1;
      const float rp2 = -z0 + s2 - h2;
      const float rp3 = -z1 + s3 - h3;
      const float rpeq = wsum(z0 + z1) - bb;
      const float rc0 = l0 * s0 - mu, rc1 = l1 * s1 - mu;
      const float rc2 = l2 * s2 - mu, rc3 = l3 * s3 - mu;
      const float t0 = (l0 * rp0 - rc0) / s0, t1 = (l1 * rp1 - rc1) / s1;
      const float t2 = (l2 * rp2 - rc2) / s2, t3 = (l3 * rp3 - rc3) / s3;
      const float rhs0 = -(rd0 + (t0 - t2));
      const float rhs1 = -(rd1 + (t1 - t3));
      const float D02 = l0 / s0 + l2 / s2;
      const float D13 = l1 / s1 + l3 / s3;

      // H = Q + diag(D) in packed lower triangle
      for (int r = 0; r < NN; ++r)
        for (int cc = lane; cc <= r; cc += 32) Hi[TRI(r, cc)] = sQ[r * NN + cc];
      __syncthreads();
      Hi[TRI(lane, lane)] += D02;
      Hi[TRI(lane + 32, lane + 32)] += D13;
      Ui[lane] = rhs0;
      Ui[lane + 32] = rhs1;
      Wi[lane] = 1.0f;
      Wi[lane + 32] = 1.0f;
      __syncthreads();

      // In-place Cholesky (right-looking), one wave per item.
      for (int k = 0; k < NN; ++k) {
        const float lkk = sqrtf(Hi[TRI(k, k)]);
        const float rinv = 1.0f / lkk;
        if (lane == 0) Hi[TRI(k, k)] = lkk;
        for (int r = k + 1 + lane; r < NN; r += 32) Hi[TRI(r, k)] *= rinv;
        __syncthreads();
        for (int j = k + 1; j < NN; ++j) {
          const float ljk = Hi[TRI(j, k)];
          for (int r = j + lane; r < NN; r += 32)
            Hi[TRI(r, j)] -= Hi[TRI(r, k)] * ljk;
        }
        __syncthreads();
      }

      // Forward: L y = rhs (two RHS in lockstep)
      for (int k = 0; k < NN; ++k) {
        const float lkk = Hi[TRI(k, k)];
        const float uk = Ui[k] / lkk;
        const float wk = Wi[k] / lkk;
        if (lane == 0) { Ui[k] = uk; Wi[k] = wk; }
        for (int r = k + 1 + lane; r < NN; r += 32) {
          const float lrk = Hi[TRI(r, k)];
          Ui[r] -= lrk * uk;
          Wi[r] -= lrk * wk;
        }
        __syncthreads();
      }
      // Backward: L^T x = y
      for (int k = NN - 1; k >= 0; --k) {
        const float lkk = Hi[TRI(k, k)];
        const float uk = Ui[k] / lkk;
        const float wk = Wi[k] / lkk;
        if (lane == 0) { Ui[k] = uk; Wi[k] = wk; }
        for (int r = lane; r < k; r += 32) {
          const float lkr = Hi[TRI(k, r)];
          Ui[r] -= lkr * uk;
          Wi[r] -= lkr * wk;
        }
        __syncthreads();
      }

      // Schur for dnu: sum(u) - sum(w)*dnu = -rpeq
      const float u0 = Ui[lane], u1 = Ui[lane + 32];
      const float w0 = Wi[lane], w1 = Wi[lane + 32];
      const float dnu = (wsum(u0 + u1) + rpeq) / wsum(w0 + w1);
      const float dz0 = u0 - dnu * w0;
      const float dz1 = u1 - dnu * w1;

      const float dl0 = (l0 * (rp0 + dz0) - rc0) / s0;
      const float dl1 = (l1 * (rp1 + dz1) - rc1) / s1;
      const float dl2 = (l2 * (rp2 - dz0) - rc2) / s2;
      const float dl3 = (l3 * (rp3 - dz1) - rc3) / s3;
      const float ds0 = -rp0 - dz0, ds1 = -rp1 - dz1;
      const float ds2 = -rp2 + dz0, ds3 = -rp3 + dz1;

      float am = 1.0f;
      am = fminf(am, dl0 < 0.f ? -l0 / dl0 : 1.0f);
      am = fminf(am, dl1 < 0.f ? -l1 / dl1 : 1.0f);
      am = fminf(am, dl2 < 0.f ? -l2 / dl2 : 1.0f);
      am = fminf(am, dl3 < 0.f ? -l3 / dl3 : 1.0f);
      am = fminf(am, ds0 < 0.f ? -s0 / ds0 : 1.0f);
      am = fminf(am, ds1 < 0.f ? -s1 / ds1 : 1.0f);
      am = fminf(am, ds2 < 0.f ? -s2 / ds2 : 1.0f);
      am = fminf(am, ds3 < 0.f ? -s3 / ds3 : 1.0f);
      const float a = 0.99f * fminf(1.0f, wmin(am));

      z0 += a * dz0; z1 += a * dz1;
      s0 += a * ds0; s1 += a * ds1; s2 += a * ds2; s3 += a * ds3;
      l0 += a * dl0; l1 += a * dl1; l2 += a * dl2; l3 += a * dl3;
      nu += a * dnu;
    }

    d0 = z0;
    d1 = z1;
    x0 += d0;  // ALPHA = 1
    x1 += d1;
  }

  // reference returns x + d (x already includes the last d)
  out[gitem * NN + lane] = x0 + d0;
  out[gitem * NN + lane + 32] = x1 + d1;
}

extern "C" void kernel_launch(void* const* d_in, const int* in_sizes, int n_in,
                              void* d_out, int out_size, void* d_ws,
                              size_t ws_size, hipStream_t stream) {
  const float* c = (const float*)d_in[0];   // (1024, 64) f32
  const float* Q = (const float*)d_in[1];   // (64, 64) f32
  float* out = (float*)d_out;               // (1024, 64) f32
  const int B = in_sizes[0] / NN;           // 1024
  sqp_kernel<<<dim3(B / ITEMS), dim3(WG), 0, stream>>>(c, Q, out);
}